// PIMLHBV_Attention_37065567764550
// MI455X (gfx1250) — compile-verified
//
#include <hip/hip_runtime.h>
#include <hip/hip_bf16.h>

typedef float v2f __attribute__((ext_vector_type(2)));
typedef float v8f __attribute__((ext_vector_type(8)));

// D = A(16x4 f32) * B(4x16 f32) + C(16x16 f32), full-precision CDNA5 WMMA.
#define WMMA_F32_K4(A, B, C) \
  __builtin_amdgcn_wmma_f32_16x16x4_f32(false, (A), false, (B), (short)0, (C), false, false)

__device__ __forceinline__ v2f mkv2(float x, float y) { v2f v; v.x = x; v.y = y; return v; }
__device__ __forceinline__ v8f splat8(float x) {
  v8f v;
#pragma unroll
  for (int i = 0; i < 8; ++i) v[i] = x;
  return v;
}
// Zero-padded weight fetch so K/N padding of the tiny MLP matrices is implicit.
__device__ __forceinline__ float wget(const float* __restrict__ W, int K, int N, int k, int n) {
  return (k < K && n < N) ? W[k * N + n] : 0.0f;
}
__device__ __forceinline__ float heav(float x) { return (tanhf(5.0f * x) + 1.0f) * 0.5f; }
__device__ __forceinline__ float clip1e5(float x) {
  return fminf(fmaxf(x, -100000.0f), 100000.0f);
}

// One wave32 per workgroup; wave handles a 16-row batch tile across all S steps.
// Lane L and L+16 both own batch row (L&15): state is mirrored in registers in
// both halves so the 16x4 f32 WMMA A-operand (lanes 0-15: K=0,1; lanes 16-31:
// K=2,3) needs no LDS traffic. Layer outputs (C layout) are transposed to the
// next layer's A layout through a small padded LDS tile (row stride 36 floats
// -> conflict-free ds_load_b64 on the A reads).
__global__ __launch_bounds__(32)
void hbv_scan_kernel(const float* __restrict__ inputs,  // (B,S,3)
                     const float* __restrict__ bp,      // (13,)
                     const float* __restrict__ W1, const float* __restrict__ b1,
                     const float* __restrict__ W2, const float* __restrict__ b2,
                     const float* __restrict__ W3, const float* __restrict__ b3,
                     float* __restrict__ qws,           // (B,S) scratch
                     int S)
{
  const int lane = threadIdx.x;   // 0..31
  const int hf   = lane >> 4;     // K-half selector for A/B operands
  const int r    = lane & 15;     // row index (A/C "M") == col index (B/C "N")
  const int row  = blockIdx.x * 16 + r;

  __shared__ float sH[16 * 36];   // h1 (16x32) then h2 (16x16) staging, padded
  __shared__ float sSC[16 * 8];   // scores (16x5) staging

  // ---- physics parameters (uniform -> SGPRs) ----
  const float Tt    = bp[0] * 4.0f - 1.5f;
  const float CFR   = bp[1] * 0.1f;
  const float CFMAX = bp[2] * 9.0f + 1.0f;
  const float SCF   = bp[3] * 0.6f + 0.4f;
  const float FC    = bp[4] * 50.0f + 50.0f;
  const float Beta  = bp[5] * 5.0f + 1.0f;
  const float LP    = bp[6] * 0.7f + 0.3f;
  const float CWH   = bp[7] * 0.2f;
  const float UZL   = bp[8] * 70.0f;
  const float K0    = bp[9] * 0.4f + 0.1f;
  const float K1    = bp[10] * 0.39f + 0.01f;
  const float K2    = bp[11] * 0.149f + 0.001f;
  const float PER   = bp[12] * 3.0f;
  const float CWHr  = bp[7];
  const float FCLP  = FC * LP;

  // ---- B-operands in WMMA f32 16x16x4 layout ----
  // lane: N = r (+16 per N-tile); VGPR pair holds K = k0+2*hf, k0+2*hf+1.
  v2f bw1[2][2];
#pragma unroll
  for (int ti = 0; ti < 2; ++ti)
#pragma unroll
    for (int c = 0; c < 2; ++c) {
      int k = 4 * c + 2 * hf, n = r + 16 * ti;
      bw1[ti][c] = mkv2(wget(W1, 5, 32, k, n), wget(W1, 5, 32, k + 1, n));
    }
  v2f bw2[8];
#pragma unroll
  for (int c = 0; c < 8; ++c) {
    int k = 4 * c + 2 * hf;
    bw2[c] = mkv2(wget(W2, 32, 16, k, r), wget(W2, 32, 16, k + 1, r));
  }
  v2f bw3[4];
#pragma unroll
  for (int c = 0; c < 4; ++c) {
    int k = 4 * c + 2 * hf;
    bw3[c] = mkv2(wget(W3, 16, 5, k, r), wget(W3, 16, 5, k + 1, r));
  }
  // biases broadcast along the C-layout column (lane) dimension
  const float bias1a = b1[r];
  const float bias1b = b1[r + 16];
  const float bias2  = b2[r];
  const float bias3  = (r < 5) ? b3[r] : 0.0f;

  float st[5] = {0.0f, 0.0f, 0.0f, 0.0f, 0.0f};
  const float* __restrict__ xrow = inputs + (size_t)row * (size_t)S * 3;

#pragma unroll 1
  for (int t = 0; t < S; ++t) {
    const float P   = xrow[3 * t + 0];
    const float T   = xrow[3 * t + 1];
    const float PET = xrow[3 * t + 2];

    // ---- layer 1: h1 = relu(S_t @ W1 + b1), K=5 padded to 8 ----
    v2f a0 = mkv2(hf ? st[2] : st[0], hf ? st[3] : st[1]);
    v2f a1 = mkv2(hf ? 0.0f : st[4], 0.0f);
    v8f c1a = splat8(bias1a);
    c1a = WMMA_F32_K4(a0, bw1[0][0], c1a);
    c1a = WMMA_F32_K4(a1, bw1[0][1], c1a);
    v8f c1b = splat8(bias1b);
    c1b = WMMA_F32_K4(a0, bw1[1][0], c1b);
    c1b = WMMA_F32_K4(a1, bw1[1][1], c1b);
#pragma unroll
    for (int v = 0; v < 8; ++v) {
      int m = v + 8 * hf;                       // C layout: lanes>=16 hold M=v+8
      sH[m * 36 + r]      = fmaxf(c1a[v], 0.0f);
      sH[m * 36 + r + 16] = fmaxf(c1b[v], 0.0f);
    }
    __syncthreads();

    // ---- layer 2: h2 = relu(h1 @ W2 + b2), K=32 ----
    v8f c2 = splat8(bias2);
#pragma unroll
    for (int c = 0; c < 8; ++c) {
      v2f a = *(const v2f*)&sH[r * 36 + 4 * c + 2 * hf];
      c2 = WMMA_F32_K4(a, bw2[c], c2);
    }
    __syncthreads();
#pragma unroll
    for (int v = 0; v < 8; ++v) sH[(v + 8 * hf) * 36 + r] = fmaxf(c2[v], 0.0f);
    __syncthreads();

    // ---- layer 3: scores = h2 @ W3 + b3, K=16, N padded 5->16 ----
    v8f c3 = splat8(bias3);
#pragma unroll
    for (int c = 0; c < 4; ++c) {
      v2f a = *(const v2f*)&sH[r * 36 + 4 * c + 2 * hf];
      c3 = WMMA_F32_K4(a, bw3[c], c3);
    }
    if (r < 5) {
#pragma unroll
      for (int v = 0; v < 8; ++v) sSC[(v + 8 * hf) * 8 + r] = c3[v];
    }
    __syncthreads();

    // ---- per-row masked softmax + HBV physics (mirrored in both halves) ----
    float sc[5];
#pragma unroll
    for (int j = 0; j < 5; ++j) sc[j] = sSC[r * 8 + j];
    float z[5], mx = -3.4e38f;
#pragma unroll
    for (int j = 0; j < 5; ++j) {
      float mk = (st[j] > 1e-9f) ? 1.0f : 0.0f;
      z[j] = sc[j] * mk + (1.0f - mk) * (-1.0e9f);
      mx = fmaxf(mx, z[j]);
    }
    float e[5], se = 0.0f;
#pragma unroll
    for (int j = 0; j < 5; ++j) { e[j] = __expf(z[j] - mx); se += e[j]; }
    const float inv = 1.0f / se;
    float Se[5];
#pragma unroll
    for (int j = 0; j < 5; ++j) Se[j] = 0.5f * (st[j] * (e[j] * inv)) + 0.5f * st[j];

    const float S1 = st[0], S2 = st[1], S3 = st[2], S4 = st[3], S5 = st[4];
    const float S1e = Se[0], S2e = Se[1], S3e = Se[2], S4e = Se[3], S5e = Se[4];
    const float hTtT = heav(Tt - T), hTTt = heav(T - Tt);
    const float Ps   = P * SCF * hTtT;
    const float R_fr = fminf((Tt - T) * CFR * CFMAX, S2e) * hTtT;
    const float Pr   = P * hTTt;
    const float M    = fminf((T - Tt) * CFMAX, S1e) * hTTt;
    const float Ipot = Pr + M;
    const float Ex   = fmaxf(0.0f, S2e + Ipot - S1e * CWH);
    const float I    = Ipot - Ex;
    const float R_e  = (I + Ex) * __powf(fmaxf(S3e, 1e-8f) / FC, Beta);
    const float hS3  = heav(S3e);
    const float E_a  = hS3 * (PET * S3e / FCLP) * heav(FCLP - S3e) + hS3 * PET * heav(S3e - FCLP);
    const float Pe   = PER * heav(S4e);
    const float q0   = (S4e - UZL) * K0 * heav(S4e - UZL);
    const float q1   = S4e * K1 * heav(S4e);
    const float q2   = S5e * K2 * heav(S5e);

    st[0] = fmaxf(S1 + clip1e5(Ps + R_fr - M), 0.0f);
    st[1] = fmaxf(fminf(S2 + clip1e5(Pr + M - I - Ex - R_fr), st[0] * CWHr), 0.0f);
    st[2] = fmaxf(S3 + clip1e5(I - E_a - R_e), 0.0f);
    st[3] = fmaxf(S4 + clip1e5(R_e - q0 - q1 - Pe), 0.0f);
    st[4] = fmaxf(S5 + clip1e5(Pe - q2), 0.0f);

    if (lane < 16) qws[(size_t)row * S + t] = q0 + q1 + q2;
  }
}

// 7-tap triangular SAME conv over time, one row per block, LDS-staged.
__global__ __launch_bounds__(256)
void conv7_kernel(const float* __restrict__ qws, float* __restrict__ out, int S)
{
  const int row = blockIdx.x;
  const int tid = threadIdx.x;
  __shared__ float s[2048 + 6];
  for (int i = tid; i < S; i += 256) s[i + 3] = qws[(size_t)row * S + i];
  if (tid < 3) { s[tid] = 0.0f; s[S + 3 + tid] = 0.0f; }
  __syncthreads();

  float win[7], ssum = 0.0f;
#pragma unroll
  for (int x = 0; x < 7; ++x) {
    float v = fmaxf(0.0f, 1.0f - fabsf((float)x - 3.0f) / (3.0f + 1e-7f));
    win[x] = v; ssum += v;
  }
  float w[7];
#pragma unroll
  for (int x = 0; x < 7; ++x) w[x] = win[x] / (ssum + 1e-7f);

  for (int i = tid; i < S; i += 256) {
    float acc = 0.0f;
#pragma unroll
    for (int d = 0; d < 7; ++d) acc = fmaf(w[d], s[i + d], acc);
    out[(size_t)row * S + i] = acc;
  }
}

extern "C" void kernel_launch(void* const* d_in, const int* in_sizes, int n_in,
                              void* d_out, int out_size, void* d_ws, size_t ws_size,
                              hipStream_t stream) {
  const float* inputs = (const float*)d_in[0];
  const float* bp     = (const float*)d_in[1];
  const float* W1     = (const float*)d_in[2];
  const float* b1     = (const float*)d_in[3];
  const float* W2     = (const float*)d_in[4];
  const float* b2     = (const float*)d_in[5];
  const float* W3     = (const float*)d_in[6];
  const float* b3     = (const float*)d_in[7];
  float* out = (float*)d_out;
  float* qws = (float*)d_ws;          // B*S floats = 4 MB scratch
  const int B = 512, S = 2048;

  hbv_scan_kernel<<<B / 16, 32, 0, stream>>>(inputs, bp, W1, b1, W2, b2, W3, b3, qws, S);
  conv7_kernel<<<B, 256, 0, stream>>>(qws, out, S);
}